// MemN2N_67791763800349
// MI455X (gfx1250) — compile-verified
//
#include <hip/hip_runtime.h>
#include <hip/hip_bf16.h>
#include <math.h>
#include <stdint.h>

typedef __attribute__((ext_vector_type(2))) float v2f;
typedef __attribute__((ext_vector_type(8))) float v8f;

#define B_ 128
#define M_ 200
#define S_ 20
#define Q_ 20
#define H_ 3
#define V_ 50000
#define D_ 128

// u[b,d] = sum_q E0[query[b,q], d]
__global__ __launch_bounds__(128) void query_embed_kernel(
    const int* __restrict__ query, const float* __restrict__ E0,
    float* __restrict__ u) {
  const int b = blockIdx.x, d = threadIdx.x;
  __shared__ int idx[Q_];
  if (threadIdx.x < Q_) idx[threadIdx.x] = query[b * Q_ + threadIdx.x];
  __syncthreads();
  float acc = 0.f;
  #pragma unroll
  for (int q = 0; q < Q_; ++q) acc += E0[(size_t)idx[q] * D_ + d];
  u[(size_t)b * D_ + d] = acc;
}

// memsum[h,b,m,d] = sum_s E[h, story[b,m,s], d] + T[h,m,d]
__global__ __launch_bounds__(128) void story_embed_kernel(
    const int* __restrict__ story, const float* __restrict__ E,
    const float* __restrict__ T, float* __restrict__ memsum) {
  const int m = blockIdx.x, b = blockIdx.y, h = blockIdx.z;
  const int d = threadIdx.x;
  __shared__ int idx[S_];
  if (threadIdx.x < S_) idx[threadIdx.x] = story[((size_t)b * M_ + m) * S_ + threadIdx.x];
  __syncthreads();
  const float* Eh = E + (size_t)h * V_ * D_;
  float acc = T[((size_t)h * M_ + m) * D_ + d];
  #pragma unroll 4
  for (int s = 0; s < S_; ++s) acc += Eh[(size_t)idx[s] * D_ + d];
  memsum[(((size_t)h * B_ + b) * M_ + m) * D_ + d] = acc;
}

// One block per batch row: uTm -> softmax over M -> o -> u += o
__global__ __launch_bounds__(256) void hop_kernel(
    const float* __restrict__ msum_h, const float* __restrict__ msum_h1,
    float* __restrict__ u) {
  const int b = blockIdx.x;
  const int t = threadIdx.x;
  __shared__ float su[D_];
  __shared__ float sp[M_];
  __shared__ float red[8];

  if (t < D_) su[t] = u[(size_t)b * D_ + t];
  __syncthreads();

  // phase 1: uTm[m] = dot(memsum_h[b,m,:], u[b,:])
  if (t < M_) {
    const float* mm = msum_h + ((size_t)b * M_ + t) * D_;
    float acc = 0.f;
    #pragma unroll 8
    for (int d = 0; d < D_; ++d) acc += mm[d] * su[d];
    sp[t] = acc;
  }
  __syncthreads();

  // softmax over sp[0..M_-1]
  float x = (t < M_) ? sp[t] : -1e30f;
  for (int off = 16; off > 0; off >>= 1) x = fmaxf(x, __shfl_down(x, off, 32));
  if ((t & 31) == 0) red[t >> 5] = x;
  __syncthreads();
  if (t == 0) { float m = red[0]; for (int i = 1; i < 8; ++i) m = fmaxf(m, red[i]); red[0] = m; }
  __syncthreads();
  const float mx = red[0];
  const float e = (t < M_) ? expf(sp[t] - mx) : 0.f;
  float s = e;
  for (int off = 16; off > 0; off >>= 1) s += __shfl_down(s, off, 32);
  __syncthreads();          // all reads of red[0] done before rewrite
  if ((t & 31) == 0) red[t >> 5] = s;
  __syncthreads();
  if (t == 0) { float ss = 0.f; for (int i = 0; i < 8; ++i) ss += red[i]; red[0] = ss; }
  __syncthreads();
  const float inv = 1.f / red[0];
  if (t < M_) sp[t] = e * inv;
  __syncthreads();

  // phase 3: o[d] = sum_m p[m] * memsum_{h+1}[b,m,d]; u += o
  if (t < D_) {
    const float* cc = msum_h1 + (size_t)b * M_ * D_ + t;
    float o = 0.f;
    #pragma unroll 8
    for (int m = 0; m < M_; ++m) o += sp[m] * cc[(size_t)m * D_];
    u[(size_t)b * D_ + t] += o;
  }
}

// ahat[b, v] = sum_k u[b,k] * E3[v,k]  via V_WMMA_F32_16X16X4_F32.
// Block: 8 waves. All waves share one 16-wide vocab tile (blockIdx.x); wave w
// owns batch rows [16w, 16w+16). The shared E3 tile (16x128 f32 = 8 KB) is
// staged into LDS once per block via GLOBAL_LOAD_ASYNC_TO_LDS_B128 (ASYNCcnt),
// removing the 8x redundant global reads of E3.
__global__ __launch_bounds__(256) void gemm_wmma_kernel(
    const float* __restrict__ u, const float* __restrict__ E3,
    float* __restrict__ ahat) {
  __shared__ float btile[16 * D_];      // 8 KB
  const int t = threadIdx.x;
  const int wave = t >> 5;
  const int lane = t & 31;
  const int vtile = blockIdx.x;

  // ---- async stage: 512 x 16B chunks, 2 per thread ----
  {
    const char* src = (const char*)(E3 + (size_t)vtile * 16 * D_);
    const uint32_t ldsbase = (uint32_t)(uintptr_t)(void*)btile; // low 32 bits = LDS offset
    #pragma unroll
    for (int i = 0; i < 2; ++i) {
      const int c = t + i * 256;
      uint32_t lds_off = ldsbase + (uint32_t)(c * 16);
      uint64_t gaddr = (uint64_t)(uintptr_t)(src + (size_t)c * 16);
      asm volatile("global_load_async_to_lds_b128 %0, %1, off"
                   :: "v"(lds_off), "v"(gaddr) : "memory");
    }
    asm volatile("s_wait_asynccnt 0x0" ::: "memory");
  }
  __syncthreads();

  const int m16 = lane & 15;
  const int khalf = (lane & 16) ? 2 : 0;   // A/B 16x4 f32 lane layout (ISA 7.12.2)

  const float* arow = u + ((size_t)(wave * 16 + m16)) * D_;
  const float* brow = btile + (size_t)m16 * D_;   // LDS-resident E3 rows

  v8f acc = {};
  #pragma unroll 4
  for (int k0 = 0; k0 < D_; k0 += 4) {
    v2f a = *(const v2f*)(arow + k0 + khalf);   // u[b, k0+khalf .. +1]  (global, cached)
    v2f b = *(const v2f*)(brow + k0 + khalf);   // E3[v, k0+khalf .. +1] (ds_load_b64)
    acc = __builtin_amdgcn_wmma_f32_16x16x4_f32(
        /*neg_a=*/false, a, /*neg_b=*/false, b,
        /*c_mod=*/(short)0, acc, /*reuse_a=*/false, /*reuse_b=*/false);
  }

  // C/D layout: lanes 0-15 -> M=j, lanes 16-31 -> M=j+8; N = lane&15
  const int nidx = vtile * 16 + m16;
  const int mbase = wave * 16 + ((lane & 16) ? 8 : 0);
  #pragma unroll
  for (int j = 0; j < 8; ++j) {
    ahat[(size_t)(mbase + j) * V_ + nidx] = acc[j];
  }
}

// Per-row softmax over V (one block per batch row)
__global__ __launch_bounds__(256) void softmax_v_kernel(
    const float* __restrict__ ahat, float* __restrict__ outp) {
  const int b = blockIdx.x;
  const int t = threadIdx.x;
  const float* row = ahat + (size_t)b * V_;
  float* orow = outp + (size_t)b * V_;
  __shared__ float red[8];

  float mx = -1e30f;
  for (int v = t; v < V_; v += 256) mx = fmaxf(mx, row[v]);
  for (int off = 16; off > 0; off >>= 1) mx = fmaxf(mx, __shfl_down(mx, off, 32));
  if ((t & 31) == 0) red[t >> 5] = mx;
  __syncthreads();
  if (t == 0) { float m = red[0]; for (int i = 1; i < 8; ++i) m = fmaxf(m, red[i]); red[0] = m; }
  __syncthreads();
  const float rmax = red[0];
  __syncthreads();

  float s = 0.f;
  for (int v = t; v < V_; v += 256) s += expf(row[v] - rmax);
  for (int off = 16; off > 0; off >>= 1) s += __shfl_down(s, off, 32);
  if ((t & 31) == 0) red[t >> 5] = s;
  __syncthreads();
  if (t == 0) { float ss = 0.f; for (int i = 0; i < 8; ++i) ss += red[i]; red[0] = ss; }
  __syncthreads();
  const float inv = 1.f / red[0];

  for (int v = t; v < V_; v += 256) orow[v] = expf(row[v] - rmax) * inv;
}

extern "C" void kernel_launch(void* const* d_in, const int* in_sizes, int n_in,
                              void* d_out, int out_size, void* d_ws, size_t ws_size,
                              hipStream_t stream) {
  (void)in_sizes; (void)n_in; (void)out_size; (void)ws_size;
  const int*   story = (const int*)d_in[0];    // [B,M,S]
  const int*   query = (const int*)d_in[1];    // [B,Q]
  const float* E     = (const float*)d_in[2];  // [H+1,V,D]
  const float* T     = (const float*)d_in[3];  // [H+1,M,D]

  // Workspace layout (floats): memsum[4,B,M,D] then u[B,D] (~52.5 MB total)
  float* memsum = (float*)d_ws;
  float* u      = memsum + (size_t)(H_ + 1) * B_ * M_ * D_;

  float* ahat = (float*)d_out;                 // [B,V]
  float* soft = ahat + (size_t)B_ * V_;        // [B,V]

  query_embed_kernel<<<B_, 128, 0, stream>>>(query, E, u);

  dim3 sgrid(M_, B_, H_ + 1);
  story_embed_kernel<<<sgrid, 128, 0, stream>>>(story, E, T, memsum);

  for (int hop = 0; hop < H_; ++hop) {
    hop_kernel<<<B_, 256, 0, stream>>>(
        memsum + (size_t)hop * B_ * M_ * D_,
        memsum + (size_t)(hop + 1) * B_ * M_ * D_, u);
  }

  gemm_wmma_kernel<<<V_ / 16, 256, 0, stream>>>(u, E + (size_t)H_ * V_ * D_, ahat);

  softmax_v_kernel<<<B_, 256, 0, stream>>>(ahat, soft);
}